// CompGCN_4896262717937
// MI455X (gfx1250) — compile-verified
//
#include <hip/hip_runtime.h>

typedef float v2f __attribute__((ext_vector_type(2)));
typedef float v8f __attribute__((ext_vector_type(8)));

#define N_NODES 50000
#define N_EDGES 800000
#define D       128

// ---------------------------------------------------------------------------
// S[n][:] = x[n][:] - rel0[:]   (self-loop term folded into the scatter buffer)
// ---------------------------------------------------------------------------
__global__ __launch_bounds__(256)
void init_s_kernel(const float* __restrict__ x, const float* __restrict__ rel0,
                   float* __restrict__ S) {
    int idx = blockIdx.x * blockDim.x + threadIdx.x;       // over N_NODES * D/4
    if (idx >= N_NODES * (D / 4)) return;
    int d4 = idx & (D / 4 - 1);
    float4 xv = ((const float4*)x)[idx];
    float4 rv = ((const float4*)rel0)[d4];
    float4 s;
    s.x = xv.x - rv.x; s.y = xv.y - rv.y;
    s.z = xv.z - rv.z; s.w = xv.w - rv.w;
    ((float4*)S)[idx] = s;
}

// ---------------------------------------------------------------------------
// For each edge e: S[dst[e]][:] += x[src[e]][:] - rel[etype[e]][:]
// One wave32 per edge, 4 floats per lane, hardware f32 atomics (L2-resident).
// ---------------------------------------------------------------------------
__global__ __launch_bounds__(256)
void edge_scatter_kernel(const float* __restrict__ x, const float* __restrict__ rel,
                         const int* __restrict__ src, const int* __restrict__ dst,
                         const int* __restrict__ etype, float* __restrict__ S) {
    int gid  = blockIdx.x * blockDim.x + threadIdx.x;
    int edge = gid >> 5;
    int lane = threadIdx.x & 31;
    if (edge >= N_EDGES) return;

    int s = src[edge];
    int d = dst[edge];
    int t = etype[edge];

    float4 xv = ((const float4*)(x   + (size_t)s * D))[lane];
    float4 rv = ((const float4*)(rel + (size_t)t * D))[lane];

    float* o = S + (size_t)d * D + lane * 4;
    unsafeAtomicAdd(o + 0, xv.x - rv.x);
    unsafeAtomicAdd(o + 1, xv.y - rv.y);
    unsafeAtomicAdd(o + 2, xv.z - rv.z);
    unsafeAtomicAdd(o + 3, xv.w - rv.w);
}

// ---------------------------------------------------------------------------
// out = S @ WI^T + X @ WO^T + bias   (optionally ReLU'd)
// One wave per 16x16 output tile; K=128 swept in steps of 4 with
// V_WMMA_F32_16X16X4_F32, two chained accumulations per step.
//
// f32 WMMA fragment layouts (ISA 7.12.2):
//   A (16x4):  lanes 0-15 -> rows, vgpr{0,1} = K{0,1}; lanes 16-31 = K{2,3}
//   B (4x16):  mirror: lanes hold N, half-wave selects K pair
//   C (16x16): vgpr r -> M = r + 8*(lane>=16), N = lane&15
// ---------------------------------------------------------------------------
template <bool RELU>
__global__ __launch_bounds__(256)
void gemm_dual_kernel(const float* __restrict__ S,    // [N_NODES, D]
                      const float* __restrict__ X,    // [N_NODES, D]
                      const float* __restrict__ WI,   // [D, D] row-major
                      const float* __restrict__ WO,   // [D, D] row-major
                      const float* __restrict__ bias, // [D]
                      float* __restrict__ out) {      // [N_NODES, D]
    const int wave = threadIdx.x >> 5;
    const int lane = threadIdx.x & 31;
    const int tile = blockIdx.x * 8 + wave;    // 3125 blocks * 8 waves = 25000 tiles
    const int rt   = tile >> 3;                // row tile: 0..3124
    const int ct   = tile & 7;                 // col tile: 0..7
    const int half = lane >> 4;                // 0 or 1
    const int l15  = lane & 15;

    // A fragments: row of S/X, K-pair selected by half-wave
    const int    arow = rt * 16 + l15;
    const float* aS   = S + (size_t)arow * D + 2 * half;
    const float* aX   = X + (size_t)arow * D + 2 * half;
    // B fragments: B[k][j] = W[j][k] -> contiguous float2 from W row j
    const int    brow = ct * 16 + l15;
    const float* bI   = WI + (size_t)brow * D + 2 * half;
    const float* bO   = WO + (size_t)brow * D + 2 * half;

    v8f c = {0.f, 0.f, 0.f, 0.f, 0.f, 0.f, 0.f, 0.f};

#pragma unroll
    for (int k = 0; k < D; k += 4) {
        v2f a0 = *(const v2f*)(aS + k);
        v2f b0 = *(const v2f*)(bI + k);
        c = __builtin_amdgcn_wmma_f32_16x16x4_f32(false, a0, false, b0,
                                                  (short)0, c, false, false);
        v2f a1 = *(const v2f*)(aX + k);
        v2f b1 = *(const v2f*)(bO + k);
        c = __builtin_amdgcn_wmma_f32_16x16x4_f32(false, a1, false, b1,
                                                  (short)0, c, false, false);
    }

    const int   col = ct * 16 + l15;
    const float bv  = bias[col];
    float* orow = out + (size_t)(rt * 16 + half * 8) * D + col;
#pragma unroll
    for (int r = 0; r < 8; ++r) {
        float v = c[r] + bv;
        if (RELU) v = fmaxf(v, 0.f);
        orow[(size_t)r * D] = v;
    }
}

// ---------------------------------------------------------------------------
extern "C" void kernel_launch(void* const* d_in, const int* in_sizes, int n_in,
                              void* d_out, int out_size, void* d_ws, size_t ws_size,
                              hipStream_t stream) {
    const float* x    = (const float*)d_in[0];
    const int*   ei   = (const int*)  d_in[1];   // [2, N_EDGES]
    const int*   et   = (const int*)  d_in[2];   // [N_EDGES]
    const float* WI1  = (const float*)d_in[3];
    const float* WO1  = (const float*)d_in[4];
    const float* rel1 = (const float*)d_in[5];
    const float* b1   = (const float*)d_in[6];
    const float* WI2  = (const float*)d_in[7];
    const float* WO2  = (const float*)d_in[8];
    const float* rel2 = (const float*)d_in[9];
    const float* b2   = (const float*)d_in[10];

    const int* src = ei;
    const int* dst = ei + N_EDGES;

    float* S = (float*)d_ws;                        // [N_NODES, D] scatter buffer
    float* h = S + (size_t)N_NODES * D;             // [N_NODES, D] layer-1 output
    float* o = (float*)d_out;                       // [N_NODES, D]

    const int initBlocks = (N_NODES * (D / 4) + 255) / 256;   // 6250
    const int edgeBlocks = (N_EDGES * 32 + 255) / 256;        // 100000
    const int gemmBlocks = N_NODES / 16 / 8 * 8 / 8;          // 390
    (void)gemmBlocks;

    // ---- layer 1: h = relu(S1 @ WI1^T + x @ WO1^T + b1), S1 = scatter + self
    init_s_kernel<<<initBlocks, 256, 0, stream>>>(x, rel1, S);
    edge_scatter_kernel<<<edgeBlocks, 256, 0, stream>>>(x, rel1, src, dst, et, S);
    gemm_dual_kernel<true><<<3125, 256, 0, stream>>>(S, x, WI1, WO1, b1, h);

    // ---- layer 2: out = S2 @ WI2^T + h @ WO2^T + b2
    init_s_kernel<<<initBlocks, 256, 0, stream>>>(h, rel2, S);
    edge_scatter_kernel<<<edgeBlocks, 256, 0, stream>>>(h, rel2, src, dst, et, S);
    gemm_dual_kernel<false><<<3125, 256, 0, stream>>>(S, h, WI2, WO2, b2, o);
}